// MolecularGCN_4595615007149
// MI455X (gfx1250) — compile-verified
//
#include <hip/hip_runtime.h>

typedef __attribute__((ext_vector_type(2))) float v2f;
typedef __attribute__((ext_vector_type(8))) float v8f;

#define F_DIM 128
#define FOUT_DIM 64

// ---------------------------------------------------------------- utilities
__global__ void fill_zero_kernel(float* __restrict__ p, long long n) {
  long long i = (long long)blockIdx.x * blockDim.x + threadIdx.x;
  long long stride = (long long)gridDim.x * blockDim.x;
  for (; i < n; i += stride) p[i] = 0.0f;
}

__global__ void degree_kernel(const int* __restrict__ recv,
                              float* __restrict__ deg, int E) {
  int e = blockIdx.x * blockDim.x + threadIdx.x;
  if (e < E) atomicAdd(&deg[recv[e]], 1.0f);
}

__global__ void invdeg_kernel(float* __restrict__ deg, int N) {
  int i = blockIdx.x * blockDim.x + threadIdx.x;
  if (i < N) deg[i] = 1.0f / fmaxf(deg[i], 1.0f);
}

// ------------------------------------------------- edge gather + scatter-add
// One wave32 per edge: each lane moves 4 consecutive floats (128 features).
__global__ __launch_bounds__(256) void scatter_kernel(
    const float* __restrict__ h, const int* __restrict__ send,
    const int* __restrict__ recv, float* __restrict__ agg, int E) {
  int wave = (int)((blockIdx.x * blockDim.x + threadIdx.x) >> 5);
  int lane = threadIdx.x & 31;
  if (wave >= E) return;
  int s = send[wave];
  int r = recv[wave];
  const float4 v =
      *reinterpret_cast<const float4*>(h + (long long)s * F_DIM + lane * 4);
  float* dst = agg + (long long)r * F_DIM + lane * 4;
  atomicAdd(dst + 0, v.x);
  atomicAdd(dst + 1, v.y);
  atomicAdd(dst + 2, v.z);
  atomicAdd(dst + 3, v.w);
}

// ------------------------------------------- [N,128] x [128,128] WMMA + ReLU
// out[n,j] = relu( (agg[n,:] * invdeg[n]) @ W + b[j] )
// Block: 256 threads = 8 waves; wave w owns output columns [16w, 16w+16).
// LDS: full W (64KB) + 16x128 A tile (8KB) + bias (512B), dynamic smem.
__global__ __launch_bounds__(256) void gcn_gemm_relu_kernel(
    const float* __restrict__ A, const float* __restrict__ invdeg,
    const float* __restrict__ W, const float* __restrict__ bias,
    float* __restrict__ out, int N) {
  extern __shared__ float smem[];
  float* sW = smem;             // 128*128
  float* sA = smem + 16384;     // 16*128
  float* sBias = smem + 18432;  // 128

  {  // stage W once per block
    const float4* Wv = reinterpret_cast<const float4*>(W);
    float4* sWv = reinterpret_cast<float4*>(sW);
    for (int i = threadIdx.x; i < 16384 / 4; i += blockDim.x) sWv[i] = Wv[i];
    if (threadIdx.x < 128) sBias[threadIdx.x] = bias[threadIdx.x];
  }
  __syncthreads();

  const int lane = threadIdx.x & 31;
  const int wv = threadIdx.x >> 5;  // n-tile 0..7
  const int hi = lane >> 4;         // lane half
  const int mn = lane & 15;         // row (A) / col (B,D) within tile

  const int numTiles = (N + 15) >> 4;
  for (int mt = blockIdx.x; mt < numTiles; mt += gridDim.x) {
    __syncthreads();
    {  // stage 16x128 A tile scaled by inv_deg
      const float4* Av = reinterpret_cast<const float4*>(A);
      float4* sAv = reinterpret_cast<float4*>(sA);
      for (int i = threadIdx.x; i < 512; i += blockDim.x) {
        int row = i >> 5;  // 32 float4 per row
        int node = (mt << 4) + row;
        float4 v = make_float4(0.f, 0.f, 0.f, 0.f);
        if (node < N) {
          v = Av[(long long)node * 32 + (i & 31)];
          float s = invdeg[node];
          v.x *= s; v.y *= s; v.z *= s; v.w *= s;
        }
        sAv[i] = v;
      }
    }
    __syncthreads();

    v8f acc = {};
    const int ncol = (wv << 4) + mn;
    for (int kk = 0; kk < 128; kk += 4) {
      const int k0 = kk + (hi << 1);
      v2f a, b;
      a.x = sA[mn * 128 + k0];
      a.y = sA[mn * 128 + k0 + 1];
      b.x = sW[k0 * 128 + ncol];
      b.y = sW[(k0 + 1) * 128 + ncol];
      acc = __builtin_amdgcn_wmma_f32_16x16x4_f32(false, a, false, b, (short)0,
                                                  acc, false, false);
    }

    const float bcol = sBias[ncol];
    for (int v = 0; v < 8; ++v) {
      int row = v + (hi << 3);
      int node = (mt << 4) + row;
      if (node < N)
        out[(long long)node * 128 + ncol] = fmaxf(acc[v] + bcol, 0.0f);
    }
  }
}

// -------------------------------------------------------- graph mean pooling
__global__ __launch_bounds__(128) void pool_kernel(const float* __restrict__ h,
                                                   float* __restrict__ pooled,
                                                   int G, int NP) {
  int g = blockIdx.x;
  int f = threadIdx.x;
  if (g >= G) return;
  float sum = 0.0f;
  const float* base = h + (long long)g * NP * F_DIM + f;
  for (int i = 0; i < NP; ++i) sum += base[(long long)i * F_DIM];
  pooled[(long long)g * F_DIM + f] = sum / (float)NP;
}

// ------------------------------------------- [G,128] x [128,64] WMMA + bias
__global__ __launch_bounds__(128) void out_gemm_kernel(
    const float* __restrict__ Ain, const float* __restrict__ W,
    const float* __restrict__ bias, float* __restrict__ out, int G) {
  extern __shared__ float smem[];
  float* sW = smem;             // 128*64
  float* sA = smem + 8192;      // 16*128
  float* sBias = smem + 10240;  // 64

  {
    const float4* Wv = reinterpret_cast<const float4*>(W);
    float4* sWv = reinterpret_cast<float4*>(sW);
    for (int i = threadIdx.x; i < 8192 / 4; i += blockDim.x) sWv[i] = Wv[i];
    if (threadIdx.x < 64) sBias[threadIdx.x] = bias[threadIdx.x];
  }
  __syncthreads();

  const int lane = threadIdx.x & 31;
  const int wv = threadIdx.x >> 5;  // n-tile 0..3
  const int hi = lane >> 4;
  const int mn = lane & 15;

  const int numTiles = (G + 15) >> 4;
  for (int mt = blockIdx.x; mt < numTiles; mt += gridDim.x) {
    __syncthreads();
    {
      const float4* Av = reinterpret_cast<const float4*>(Ain);
      float4* sAv = reinterpret_cast<float4*>(sA);
      for (int i = threadIdx.x; i < 512; i += blockDim.x) {
        int row = i >> 5;
        int g = (mt << 4) + row;
        float4 v = make_float4(0.f, 0.f, 0.f, 0.f);
        if (g < G) v = Av[(long long)g * 32 + (i & 31)];
        sAv[i] = v;
      }
    }
    __syncthreads();

    v8f acc = {};
    const int ncol = (wv << 4) + mn;
    for (int kk = 0; kk < 128; kk += 4) {
      const int k0 = kk + (hi << 1);
      v2f a, b;
      a.x = sA[mn * 128 + k0];
      a.y = sA[mn * 128 + k0 + 1];
      b.x = sW[k0 * FOUT_DIM + ncol];
      b.y = sW[(k0 + 1) * FOUT_DIM + ncol];
      acc = __builtin_amdgcn_wmma_f32_16x16x4_f32(false, a, false, b, (short)0,
                                                  acc, false, false);
    }

    const float bcol = sBias[ncol];
    for (int v = 0; v < 8; ++v) {
      int row = v + (hi << 3);
      int g = (mt << 4) + row;
      if (g < G) out[(long long)g * FOUT_DIM + ncol] = acc[v] + bcol;
    }
  }
}

// --------------------------------------------------------------------- launch
extern "C" void kernel_launch(void* const* d_in, const int* in_sizes, int n_in,
                              void* d_out, int out_size, void* d_ws,
                              size_t ws_size, hipStream_t stream) {
  (void)n_in; (void)out_size; (void)ws_size;
  const float* nodes = (const float*)d_in[0];
  const int* senders = (const int*)d_in[1];
  const int* receivers = (const int*)d_in[2];
  const float* W1 = (const float*)d_in[4];
  const float* b1 = (const float*)d_in[5];
  const float* W2 = (const float*)d_in[6];
  const float* b2 = (const float*)d_in[7];
  const float* Wout = (const float*)d_in[8];
  const float* bout = (const float*)d_in[9];
  float* out = (float*)d_out;

  const int N = in_sizes[0] / F_DIM;
  const int E = in_sizes[1];
  const int G = in_sizes[3];
  const int NP = N / G;  // nodes per graph (constant in reference)

  // workspace layout (floats)
  float* inv_deg = (float*)d_ws;
  float* buf0 = inv_deg + (((long long)N + 255) & ~255LL);  // agg buffer
  float* buf1 = buf0 + (long long)N * F_DIM;                // hidden buffer
  float* pooled = buf1 + (long long)N * F_DIM;              // [G,128]

  const size_t smem1 = (16384 + 2048 + 128) * sizeof(float);
  const size_t smem2 = (8192 + 2048 + 64) * sizeof(float);
  const int scatterBlocks = (E * 32 + 255) / 256;

  // degree -> inv_deg (in place)
  fill_zero_kernel<<<1024, 256, 0, stream>>>(inv_deg, N);
  degree_kernel<<<(E + 255) / 256, 256, 0, stream>>>(receivers, inv_deg, E);
  invdeg_kernel<<<(N + 255) / 256, 256, 0, stream>>>(inv_deg, N);

  // layer 1
  fill_zero_kernel<<<2048, 256, 0, stream>>>(buf0, (long long)N * F_DIM);
  scatter_kernel<<<scatterBlocks, 256, 0, stream>>>(nodes, senders, receivers,
                                                    buf0, E);
  gcn_gemm_relu_kernel<<<1024, 256, smem1, stream>>>(buf0, inv_deg, W1, b1,
                                                     buf1, N);
  // layer 2
  fill_zero_kernel<<<2048, 256, 0, stream>>>(buf0, (long long)N * F_DIM);
  scatter_kernel<<<scatterBlocks, 256, 0, stream>>>(buf1, senders, receivers,
                                                    buf0, E);
  gcn_gemm_relu_kernel<<<1024, 256, smem1, stream>>>(buf0, inv_deg, W2, b2,
                                                     buf1, N);

  // pooling + output projection
  pool_kernel<<<G, 128, 0, stream>>>(buf1, pooled, G, NP);
  out_gemm_kernel<<<(G + 15) / 16, 128, smem2, stream>>>(pooled, Wout, bout,
                                                         out, G);
}